// FSQ_77103252898375
// MI455X (gfx1250) — compile-verified
//
#include <hip/hip_runtime.h>

typedef __attribute__((ext_vector_type(2))) float v2f;
typedef __attribute__((ext_vector_type(8))) float v8f;

#define D_MODEL   1024
#define NLEV      8
#define FSQ_EPS   1e-5f

#define WAVES_PB  4                      // waves per block
#define KC        32                     // features per staged chunk
#define NCHUNK    (D_MODEL / KC)         // 32 chunks
#define AROW      (KC + 4)               // padded LDS row stride (floats) -> 64-bank conflict-free
#define ABUF_F    (16 * AROW)            // floats per A buffer (16 token rows)

#define NBLOCKS   512
#define TILES_TOT 4096                   // 65536 tokens / 16
#define WAVE_SLOTS (NBLOCKS * WAVES_PB)  // 2048
#define TILES_PW  (TILES_TOT / WAVE_SLOTS) // 2 tiles per wave

// ---------------------------------------------------------------------------
// Fully fused FSQ kernel:
//   logits = rstd * (X @ (gamma.*W)^T - mu*s) + t ; out = rint(4*tanh(logits))
// - X tile (16 tokens x 32 feats) async-DMA'd to LDS (global_load_async_to_lds_b128,
//   double buffered, ASYNCcnt), then gathered into WMMA-A layout via ds_load_2addr_b64.
// - gamma-folded W pre-staged in LDS in exact WMMA-B fragment order.
// - WMMA f32 16x16x4, 4 interleaved accumulators; LN stats fused on the A stream
//   with packed (v_pk_add/v_pk_fma) accumulators, horizontal-reduced once per tile.
// ---------------------------------------------------------------------------
__global__ void __launch_bounds__(WAVES_PB * 32) fsq_fused(
    const float* __restrict__ X,
    const float* __restrict__ gamma,
    const float* __restrict__ beta,
    const float* __restrict__ W,
    float* __restrict__ out) {
  __shared__ v2f   sWgP[NCHUNK * 8 * 16];            // [kk][hi][n] B fragments, 32 KB
  __shared__ float sA[WAVES_PB][2][ABUF_F];          // per-wave double buffers, 18 KB
  __shared__ float sS[8], sT[8];                     // LayerNorm fold constants

  const int tid  = threadIdx.x;
  const int lane = tid & 31;
  const int wave = tid >> 5;

  // ---- stage gamma.*W into B-fragment order: pair p = kk*16 + hi*8 + n ----
  for (int p = tid; p < NCHUNK * 8 * 16; p += WAVES_PB * 32) {
    const int n  = p & 7;
    const int hh = (p >> 3) & 1;
    const int kk = p >> 4;
    const int k  = kk * 4 + 2 * hh;
    v2f w;
    w.x = gamma[k]     * W[n * D_MODEL + k];
    w.y = gamma[k + 1] * W[n * D_MODEL + k + 1];
    sWgP[p] = w;
  }
  // ---- s_n = sum(gamma.*W[n]), t_n = sum(beta.*W[n]); 2 rows per wave -----
  for (int rep = 0; rep < 2; ++rep) {
    const int n = wave * 2 + rep;
    float s = 0.f, t = 0.f;
    for (int d = lane; d < D_MODEL; d += 32) {
      const float w = W[n * D_MODEL + d];
      s = fmaf(gamma[d], w, s);
      t = fmaf(beta[d],  w, t);
    }
    for (int off = 16; off > 0; off >>= 1) {
      s += __shfl_down(s, off, 32);
      t += __shfl_down(t, off, 32);
    }
    if (lane == 0) { sS[n] = s; sT[n] = t; }
  }
  __syncthreads();

  const int col  = lane & 15;              // B/D column
  const int hi   = lane >> 4;              // K-half selector
  const int row0 = lane >> 3;              // staging: row group 0..3
  const unsigned colb = (unsigned)(lane & 7) * 16u;   // staging: column byte

  const float sN = sS[lane & 7];
  const float tN = sT[lane & 7];

  // LDS byte address of this wave's A buffers (flat-pointer low 32 bits == LDS offset)
  const unsigned aLds0     = (unsigned)(unsigned long long)&sA[wave][0][0];
  const unsigned bufStride = ABUF_F * 4u;
  const unsigned rowG = row0 * (D_MODEL * 4u) + colb;     // global: row-group base
  const unsigned rowL = row0 * (AROW * 4u) + colb;        // LDS:    row-group base

  for (int t = 0; t < TILES_PW; ++t) {
    const int tile = blockIdx.x * WAVES_PB + wave + t * WAVE_SLOTS;
    const int tok0 = tile * 16;
    const unsigned gtile = (unsigned)tok0 * (D_MODEL * 4u);

    // ---- prologue: async-stage chunk 0 into buffer 0 (4 x b128 per lane = 2 KB)
    {
      const unsigned g = gtile + rowG;
      const unsigned l = aLds0 + rowL;
      #pragma unroll
      for (int j = 0; j < 4; ++j) {
        asm volatile("global_load_async_to_lds_b128 %0, %1, %2"
                     :: "v"(l + (unsigned)j * (4u * AROW * 4u)),
                        "v"(g + (unsigned)j * (4u * D_MODEL * 4u)),
                        "s"(X) : "memory");
      }
    }

    v8f acc[4] = {};
    v2f sx0 = {0.f, 0.f}, sx1 = {0.f, 0.f};   // packed LN-stat accumulators
    v2f sq0 = {0.f, 0.f}, sq1 = {0.f, 0.f};   // (two pairs to break dep chains)

    for (int c = 0; c < NCHUNK; ++c) {
      const unsigned bufC = (unsigned)(c & 1);
      if (c + 1 < NCHUNK) {     // prefetch next chunk into the other buffer
        const unsigned g = gtile + (unsigned)(c + 1) * (KC * 4u) + rowG;
        const unsigned l = aLds0 + (bufC ^ 1u) * bufStride + rowL;
        #pragma unroll
        for (int j = 0; j < 4; ++j) {
          asm volatile("global_load_async_to_lds_b128 %0, %1, %2"
                       :: "v"(l + (unsigned)j * (4u * AROW * 4u)),
                          "v"(g + (unsigned)j * (4u * D_MODEL * 4u)),
                          "s"(X) : "memory");
        }
        asm volatile("s_wait_asynccnt 0x4" ::: "memory");  // chunk c landed (in-order)
      } else {
        asm volatile("s_wait_asynccnt 0x0" ::: "memory");
      }

      const float* aBuf = &sA[wave][bufC][0];
      #pragma unroll
      for (int i = 0; i < 8; ++i) {   // 8 WMMA K-steps per chunk
        const v2f a = *(const v2f*)(aBuf + (lane & 15) * AROW + i * 4 + 2 * hi);
        const v2f b = sWgP[(c * 8 + i) * 16 + hi * 8 + (lane & 7)];
        acc[i & 3] = __builtin_amdgcn_wmma_f32_16x16x4_f32(
            false, a, false, b, (short)0, acc[i & 3], false, false);
        if (i & 1) { sx1 = sx1 + a;  sq1 = a * a + sq1; }   // v_pk_add / v_pk_fma
        else       { sx0 = sx0 + a;  sq0 = a * a + sq0; }
      }
    }

    v8f accT = (acc[0] + acc[1]) + (acc[2] + acc[3]);
    // horizontal reduce packed stats, then merge complementary K-halves (lane^16)
    const v2f sxv = sx0 + sx1;
    const v2f sqv = sq0 + sq1;
    const float sx = sxv.x + sxv.y;
    const float sq = sqv.x + sqv.y;
    const float sxT = sx + __shfl_xor(sx, 16, 32);
    const float sqT = sq + __shfl_xor(sq, 16, 32);
    const float mu   = sxT * (1.f / (float)D_MODEL);
    const float var  = fmaf(-mu, mu, sqT * (1.f / (float)D_MODEL));
    const float rstd = rsqrtf(var + FSQ_EPS);

    // D layout: VGPR r, lane n (+16) -> (M = r (+8), N = n)
    #pragma unroll
    for (int r = 0; r < 8; ++r) {
      const int src = r + (hi << 3);     // lane holding stats of this token row
      const float m  = __shfl(mu,   src, 32);
      const float rs = __shfl(rstd, src, 32);
      const float logit = fmaf(rs, accT[r] - m * sN, tN);
      const float q = rintf(4.f * tanhf(logit));  // round-half-even == jnp.round
      if (col < NLEV) {
        out[(size_t)(tok0 + r + (hi << 3)) * NLEV + col] = q;
      }
    }
  }
}

// ---------------------------------------------------------------------------
extern "C" void kernel_launch(void* const* d_in, const int* in_sizes, int n_in,
                              void* d_out, int out_size, void* d_ws, size_t ws_size,
                              hipStream_t stream) {
  const float* regrs = (const float*)d_in[0];   // [8,8192,1024]
  const float* gamma = (const float*)d_in[1];   // [1024]
  const float* beta  = (const float*)d_in[2];   // [1024]
  const float* W     = (const float*)d_in[3];   // [8,1024]
  float* out = (float*)d_out;                   // [8,8192,8]
  (void)d_ws; (void)ws_size;

  fsq_fused<<<NBLOCKS, WAVES_PB * 32, 0, stream>>>(regrs, gamma, beta, W, out);
}